// EdgeAttention_25546465477034
// MI455X (gfx1250) — compile-verified
//
#include <hip/hip_runtime.h>
#include <math.h>

#define N_NODES 4096
#define N_EDGES 32768
#define DV 128
#define DE 64
#define DU 32
#define DA 64
#define NEG_SLOPE 0.01f

typedef __attribute__((ext_vector_type(2))) float v2f;
typedef __attribute__((ext_vector_type(8))) float v8f;

// Monotone float -> unsigned mapping so segment-max can use atomicMax(u32).
__device__ __forceinline__ unsigned ordKey(float f) {
    unsigned u = __float_as_uint(f);
    return (u & 0x80000000u) ? ~u : (u | 0x80000000u);
}
__device__ __forceinline__ float ordInv(unsigned v) {
    return __uint_as_float((v & 0x80000000u) ? (v & 0x7FFFFFFFu) : ~v);
}

// Projection GEMM via V_WMMA_F32_16X16X4_F32, fused bias + leaky-ReLU.
// MODE 0: q rows = nodes[i]   | u[batch[i]]      (DIN = 160)
// MODE 1: k rows = nodes[s[e]]| u[batch[s[e]]]   (DIN = 160)
// MODE 2: v rows = edges[e]   | u[batch[s[e]]]   (DIN = 96)
// One wave computes one 16x16 output tile; 8 waves per 256-thread block.
template <int DIN, int MODE>
__global__ void proj_wmma_kernel(const float* __restrict__ nodes,
                                 const float* __restrict__ edges,
                                 const int* __restrict__ s,
                                 const int* __restrict__ batch,
                                 const float* __restrict__ u,
                                 const float* __restrict__ W,     // [DIN, 64]
                                 const float* __restrict__ bias,  // [64]
                                 float* __restrict__ out,         // [rows, 64]
                                 int rows) {
    const int lane = threadIdx.x & 31;
    const int wave = threadIdx.x >> 5;
    const int tile = blockIdx.x * 8 + wave;
    const int rowBase = (tile >> 2) * 16;   // 64/16 = 4 column tiles
    const int colBase = (tile & 3) * 16;
    if (rowBase >= rows) return;            // wave-uniform; EXEC stays all-1s

    const int mrow = rowBase + (lane & 15); // A-matrix row for this lane
    const int col  = colBase + (lane & 15); // B/C/D column for this lane
    const int koff = (lane >> 4) * 2;       // lanes 16-31 hold K+2, K+3

    const float* __restrict__ arow;
    int ubase;
    if (MODE == 0) {
        arow = nodes + (size_t)mrow * DV;
        ubase = batch[mrow] * DU;
    } else if (MODE == 1) {
        int sn = s[mrow];
        arow = nodes + (size_t)sn * DV;
        ubase = batch[sn] * DU;
    } else {
        arow = edges + (size_t)mrow * DE;
        ubase = batch[s[mrow]] * DU;
    }
    const int F = (MODE == 2) ? DE : DV;    // first-segment width of the concat

    v8f acc = {};
#pragma unroll
    for (int kb = 0; kb < DIN; kb += 4) {
        const int k0 = kb + koff;
        v2f a, b;
        a.x = (k0 < F)     ? arow[k0]     : u[ubase + k0 - F];
        a.y = (k0 + 1 < F) ? arow[k0 + 1] : u[ubase + k0 + 1 - F];
        b.x = W[(size_t)k0 * DA + col];
        b.y = W[(size_t)(k0 + 1) * DA + col];
        acc = __builtin_amdgcn_wmma_f32_16x16x4_f32(
            /*neg_a=*/false, a, /*neg_b=*/false, b,
            /*c_mod=*/(short)0, acc, /*reuse_a=*/false, /*reuse_b=*/false);
    }

    const float bc = bias[col];
    const int rOut = rowBase + (lane >> 4) * 8;   // lanes 16-31 hold M=8..15
#pragma unroll
    for (int j = 0; j < 8; ++j) {
        float x = acc[j] + bc;
        x = (x > 0.f) ? x : x * NEG_SLOPE;
        out[(size_t)(rOut + j) * DA + col] = x;
    }
}

__global__ void init_kernel(float* __restrict__ out, float* __restrict__ denom,
                            unsigned* __restrict__ maxOrd) {
    int i = blockIdx.x * blockDim.x + threadIdx.x;
    if (i < N_NODES * DA) out[i] = 0.f;
    if (i < N_NODES) { denom[i] = 0.f; maxOrd[i] = 0u; }
}

// Per-edge logit t[e] = dot(q[r[e]], k[e]) / 8, plus segment atomicMax.
__global__ void logit_kernel(const float* __restrict__ q, const float* __restrict__ k,
                             const int* __restrict__ recv, float* __restrict__ t,
                             unsigned* __restrict__ maxOrd) {
    int e = blockIdx.x * blockDim.x + threadIdx.x;
    if (e >= N_EDGES) return;
    int rn = recv[e];
    const float4* qf = (const float4*)(q + (size_t)rn * DA);
    const float4* kf = (const float4*)(k + (size_t)e * DA);
    float sum = 0.f;
#pragma unroll
    for (int i = 0; i < DA / 4; ++i) {
        float4 a = qf[i], b = kf[i];
        sum += a.x * b.x + a.y * b.y + a.z * b.z + a.w * b.w;
    }
    float tv = sum * 0.125f;  // 1/sqrt(64)
    t[e] = tv;
    atomicMax(maxOrd + rn, ordKey(tv));
}

// One wave per edge: w = exp(t - m[seg]); scatter-add w*v into out, w into denom.
__global__ void scatter_kernel(const float* __restrict__ t, const float* __restrict__ v,
                               const int* __restrict__ recv,
                               const unsigned* __restrict__ maxOrd,
                               float* __restrict__ denom, float* __restrict__ acc) {
    int gid = blockIdx.x * blockDim.x + threadIdx.x;
    int e = gid >> 5;
    int lane = gid & 31;
    if (e >= N_EDGES) return;
    int rn = recv[e];
    float w = __expf(t[e] - ordInv(maxOrd[rn]));
    if (lane == 0) atomicAdd(denom + rn, w);
#pragma unroll
    for (int d = lane; d < DA; d += 32)
        atomicAdd(acc + (size_t)rn * DA + d, w * v[(size_t)e * DA + d]);
}

__global__ void finalize_kernel(float* __restrict__ out, const float* __restrict__ denom) {
    int i = blockIdx.x * blockDim.x + threadIdx.x;
    if (i >= N_NODES * DA) return;
    out[i] = out[i] / denom[i / DA];
}

extern "C" void kernel_launch(void* const* d_in, const int* in_sizes, int n_in,
                              void* d_out, int out_size, void* d_ws, size_t ws_size,
                              hipStream_t stream) {
    const float* nodes = (const float*)d_in[0];
    const float* edges = (const float*)d_in[1];
    const int* edge_index = (const int*)d_in[2];
    const float* u = (const float*)d_in[3];
    const int* batch = (const int*)d_in[4];
    const float* Wq = (const float*)d_in[5];
    const float* bq = (const float*)d_in[6];
    const float* Wk = (const float*)d_in[7];
    const float* bk = (const float*)d_in[8];
    const float* Wv = (const float*)d_in[9];
    const float* bv = (const float*)d_in[10];
    const int* s = edge_index;            // edge_index[0, :]
    const int* r = edge_index + N_EDGES;  // edge_index[1, :]
    float* out = (float*)d_out;           // used directly as the accumulator

    char* ws = (char*)d_ws;
    float* q  = (float*)ws;            ws += (size_t)N_NODES * DA * 4;
    float* kk = (float*)ws;            ws += (size_t)N_EDGES * DA * 4;
    float* vv = (float*)ws;            ws += (size_t)N_EDGES * DA * 4;
    float* t  = (float*)ws;            ws += (size_t)N_EDGES * 4;
    unsigned* maxOrd = (unsigned*)ws;  ws += (size_t)N_NODES * 4;
    float* denom = (float*)ws;         ws += (size_t)N_NODES * 4;

    const int initN = N_NODES * DA;
    init_kernel<<<(initN + 255) / 256, 256, 0, stream>>>(out, denom, maxOrd);

    // tiles = (rows/16) * (64/16); 8 waves (tiles) per 256-thread block
    proj_wmma_kernel<160, 0><<<(N_NODES / 16 * 4) / 8, 256, 0, stream>>>(
        nodes, edges, s, batch, u, Wq, bq, q, N_NODES);
    proj_wmma_kernel<160, 1><<<(N_EDGES / 16 * 4) / 8, 256, 0, stream>>>(
        nodes, edges, s, batch, u, Wk, bk, kk, N_EDGES);
    proj_wmma_kernel<96, 2><<<(N_EDGES / 16 * 4) / 8, 256, 0, stream>>>(
        nodes, edges, s, batch, u, Wv, bv, vv, N_EDGES);

    logit_kernel<<<(N_EDGES + 255) / 256, 256, 0, stream>>>(q, kk, r, t, maxOrd);
    scatter_kernel<<<(N_EDGES * 32 + 255) / 256, 256, 0, stream>>>(t, vv, r, maxOrd,
                                                                   denom, out);
    finalize_kernel<<<(N_NODES * DA + 255) / 256, 256, 0, stream>>>(out, denom);
}